// LengthConverter_48945447305546
// MI455X (gfx1250) — compile-verified
//
#include <hip/hip_runtime.h>
#include <hip/hip_bf16.h>

// LengthConverter for MI455X (gfx1250).
// B=8, Lz=1024, H=1024, L_L=1024.
// weight = softmax_z(-(z-mu)^2/(2 sigma^2)) is analytically banded (sigma=1):
// exp underflows f32 beyond |d|~13.3, masked (-99) terms are ~1e-43, so each
// output row touches <= ~68 z-rows. We compute the exp band in LDS once per
// 16-row l-tile and run the banded GEMM with V_WMMA_F32_16X16X4_F32 (full f32
// precision; problem is memory/latency bound after band sparsity: ~1.4 GF vs
// 64 MB @ 23.3 TB/s).
//
// Round-2 changes: all global accesses use uniform-base + 32-bit per-lane
// offsets (GVS addressing, one v_add_nc_u32 per k-step instead of 64-bit VGPR
// address math), and the K band is aligned so k_lo + kband <= Lz, removing the
// per-iteration row clamp from the WMMA loop.

#define BB   8
#define LZ   1024
#define HD   1024
#define LL   1024
#define KMAX 96      // max K-band width (worst case ~68)
#define SROW 98      // LDS row stride in floats: keeps 8B alignment, spreads banks

typedef float v2f __attribute__((ext_vector_type(2)));
typedef float v8f __attribute__((ext_vector_type(8)));

// ---------------------------------------------------------------------------
// Phase A: n[b] = sum(z_mask[b,:]);  zp_mask output = (l < ls[b])
// ---------------------------------------------------------------------------
__global__ __launch_bounds__(256)
void lc_prep(const float* __restrict__ z_mask,
             const int* __restrict__ ls,
             float* __restrict__ n_ws,
             float* __restrict__ zp_mask_out) {
    const int b = blockIdx.x;
    __shared__ float s_sum;
    if (threadIdx.x == 0) s_sum = 0.0f;
    __syncthreads();

    float part = 0.0f;
    for (int i = threadIdx.x; i < LZ; i += blockDim.x)
        part += z_mask[b * LZ + i];
    atomicAdd(&s_sum, part);
    __syncthreads();
    if (threadIdx.x == 0) n_ws[b] = s_sum;

    const int lsb = ls[b];
    for (int i = threadIdx.x; i < LL; i += blockDim.x)
        zp_mask_out[b * LL + i] = (i < lsb) ? 1.0f : 0.0f;
}

// ---------------------------------------------------------------------------
// Phase B: banded weight build (LDS) + WMMA f32 GEMM + fused normalize/mask.
// Grid: (64 l-tiles, 8 batches), 256 threads = 8 wave32 per WG.
// Wave w handles h-tiles w, w+8, ..., w+56 (16 cols each).
// ---------------------------------------------------------------------------
__global__ __launch_bounds__(256)
void lc_main(const float* __restrict__ z,
             const int* __restrict__ ls,
             const float* __restrict__ sigma,
             const float* __restrict__ n_ws,
             float* __restrict__ out) {
    __shared__ float wexp[16 * SROW];   // un-normalized exp band, row stride SROW
    __shared__ float rowsum[16];
    __shared__ float scale[16];         // zp_mask(l) / rowsum(l)

    const int l0   = blockIdx.x * 16;   // l-tile base
    const int b    = blockIdx.y;
    const int lsb  = ls[b];
    const int tid  = threadIdx.x;
    const int wave = tid >> 5;
    const int lane = tid & 31;
    const int hi   = lane >> 4;         // 0/1: lane half
    const int lo   = lane & 15;

    float* __restrict__ outb = out + (size_t)b * LL * HD;   // uniform base

    // Whole tile masked (l >= ls): write zeros, uniform across the block.
    if (l0 >= lsb) {
        for (int ht = wave; ht < HD / 16; ht += 8) {
            const unsigned o0 = (unsigned)((l0 + 8 * hi) * HD + ht * 16 + lo);
            #pragma unroll
            for (int r = 0; r < 8; ++r)
                outb[o0 + (unsigned)(r * HD)] = 0.0f;
        }
        return;
    }

    const float nf     = n_ws[b];
    const int   nvalid = (int)nf;
    const float rr     = nf / (float)lsb;       // mu slope, in [0.5, 2]
    const float sg     = sigma[0];
    const float inv2s2 = 0.5f / (sg * sg);

    // K band covering all non-underflowing weights of rows l0..l0+15.
    // Both ends 4-aligned and k_hi <= LZ, so no load clamping is ever needed.
    int k_lo = (int)floorf((float)l0 * rr) - 16;
    if (k_lo < 0) k_lo = 0;
    k_lo &= ~3;                                  // <= 1004, 4-aligned
    int k_hi = ((int)floorf((float)(l0 + 15) * rr) + 18 + 3) & ~3;
    if (k_hi > LZ) k_hi = LZ;                    // LZ is 4-aligned
    int kband = k_hi - k_lo;
    if (kband > KMAX) kband = KMAX;
    if (kband < 4) kband = 4;

    if (tid < 16) rowsum[tid] = 0.0f;
    __syncthreads();

    // Fill 16 x KMAX exp band: 16 threads per row, 6 columns each.
    {
        const int   row = tid >> 4;
        const int   c0  = tid & 15;
        const float mu  = (float)(l0 + row) * rr;
        float part = 0.0f;
        for (int c = c0; c < KMAX; c += 16) {
            const int   k = k_lo + c;
            const float d = (float)k - mu;
            const float w = (k < nvalid) ? __expf(-(d * d) * inv2s2) : 0.0f;
            wexp[row * SROW + c] = w;
            part += w;
        }
        atomicAdd(&rowsum[row], part);          // ds_add_f32
    }
    __syncthreads();

    if (tid < 16) {
        const float s = rowsum[tid];
        scale[tid] = ((l0 + tid) < lsb && s > 0.0f) ? (1.0f / s) : 0.0f;
    }
    __syncthreads();

    // A-fragment base for this lane: row m = lo, k offset 2*hi (ISA 7.12.2).
    const float* aptr = &wexp[lo * SROW + 2 * hi];
    const float* __restrict__ zb = z + (size_t)b * LZ * HD;  // uniform base

    for (int ht = wave; ht < HD / 16; ht += 8) {
        const int h0 = ht * 16;
        v8f acc = {0.0f, 0.0f, 0.0f, 0.0f, 0.0f, 0.0f, 0.0f, 0.0f};

        // 32-bit per-lane offset; advances by 4 rows per step (v_add_nc_u32).
        unsigned off = (unsigned)((k_lo + 2 * hi) * HD + h0 + lo);

        #pragma unroll 2
        for (int kk = 0; kk < kband; kk += 4) {
            v2f a;
            a.x = aptr[kk];
            a.y = aptr[kk + 1];

            v2f bf;                              // B rows K = 2*hi, 2*hi+1
            bf.x = zb[off];
            bf.y = zb[off + (unsigned)HD];
            off += (unsigned)(4 * HD);

            // D = A(16x4) * B(4x16) + C, f32
            acc = __builtin_amdgcn_wmma_f32_16x16x4_f32(
                false, a, false, bf, (short)0, acc, false, false);
        }

        // Epilogue: normalize by rowsum and apply zp_mask (fused in scale[]).
        const unsigned o0 = (unsigned)((l0 + 8 * hi) * HD + h0 + lo);
        #pragma unroll
        for (int r = 0; r < 8; ++r)              // D vgpr r holds row r + 8*hi
            outb[o0 + (unsigned)(r * HD)] = acc[r] * scale[r + 8 * hi];
    }
}

// ---------------------------------------------------------------------------
extern "C" void kernel_launch(void* const* d_in, const int* in_sizes, int n_in,
                              void* d_out, int out_size, void* d_ws, size_t ws_size,
                              hipStream_t stream) {
    const float* z      = (const float*)d_in[0];   // [8,1024,1024] f32
    const int*   ls     = (const int*)  d_in[1];   // [8] int
    const float* z_mask = (const float*)d_in[2];   // [8,1024] f32
    const float* sigma  = (const float*)d_in[3];   // [1] f32

    float* out_zprime = (float*)d_out;                         // 8*1024*1024
    float* out_zpmask = (float*)d_out + (size_t)BB * LL * HD;  // 8*1024
    float* n_ws       = (float*)d_ws;                          // 8 floats

    lc_prep<<<BB, 256, 0, stream>>>(z_mask, ls, n_ws, out_zpmask);
    lc_main<<<dim3(LL / 16, BB), 256, 0, stream>>>(z, ls, sigma, n_ws, out_zprime);
}